// SMSTM_38405597561130
// MI455X (gfx1250) — compile-verified
//
#include <hip/hip_runtime.h>

// SOM-style fused op: norms2 = x2 + k2 - 2*x@kernel ; per-row argmin ; radial*norms2
// M=4096, K=256, N=16384, all f32. f32 WMMA (16x16x4) keeps argmin exact-ish.

typedef float v2f __attribute__((ext_vector_type(2)));
typedef float v8f __attribute__((ext_vector_type(8)));

#define M_DIM 4096
#define K_DIM 256
#define N_DIM 16384
#define MBLK 128
#define NBLK 128
#define KC 32
#define LDA 34   // padded LDS stride (even -> 8B-aligned pair loads at even k)

__device__ __forceinline__ unsigned long long som_pack(float v, unsigned n) {
  unsigned b = __float_as_uint(v);
  unsigned mono = (b & 0x80000000u) ? ~b : (b | 0x80000000u); // order-preserving
  return (((unsigned long long)mono) << 32) | (unsigned long long)n;
}

__device__ __forceinline__ unsigned long long som_shfl_xor_u64(unsigned long long v, int m) {
  unsigned lo = __shfl_xor((unsigned)v, m, 32);
  unsigned hi = __shfl_xor((unsigned)(v >> 32), m, 32);
  return (((unsigned long long)hi) << 32) | (unsigned long long)lo;
}

__global__ void som_init_min(unsigned long long* __restrict__ gmin) {
  int t = blockIdx.x * blockDim.x + threadIdx.x;
  if (t < M_DIM) gmin[t] = 0xFFFFFFFFFFFFFFFFull;
}

// row sums of x*x : one wave per row
__global__ void som_x2(const float* __restrict__ x, float* __restrict__ x2w) {
  int wave = threadIdx.x >> 5;
  int lane = threadIdx.x & 31;
  int row  = blockIdx.x * 8 + wave;
  const float* p = x + (size_t)row * K_DIM;
  float s = 0.f;
#pragma unroll
  for (int i = 0; i < 8; ++i) { float v = p[lane + 32 * i]; s += v * v; }
#pragma unroll
  for (int m = 16; m >= 1; m >>= 1) s += __shfl_xor(s, m, 32);
  if (lane == 0) x2w[row] = s;
}

// column sums of kernel*kernel : one thread per column (coalesced)
__global__ void som_k2(const float* __restrict__ kern, float* __restrict__ k2w) {
  int col = blockIdx.x * blockDim.x + threadIdx.x;
  float s = 0.f;
  for (int k = 0; k < K_DIM; ++k) { float v = kern[(size_t)k * N_DIM + col]; s += v * v; }
  k2w[col] = s;
}

// Block tile 128(M) x 128(N); 8 waves arranged 4(M) x 2(N); each wave 32x64:
// 2(M-sub) x 4(N-sub) = 8 f32 WMMA accumulators -> 8 WMMA per 6 LDS fragment loads.
__global__ __launch_bounds__(256) void som_gemm(const float* __restrict__ x,
                                                const float* __restrict__ kern,
                                                const float* __restrict__ x2w,
                                                const float* __restrict__ k2w,
                                                float* __restrict__ out,
                                                unsigned long long* __restrict__ gmin) {
  __shared__ float As[MBLK * LDA];                 // [m][k]
  __shared__ float Bs[NBLK * LDA];                 // [n][k]  (transposed on load)
  __shared__ unsigned long long minbuf[MBLK];

  const int tid  = threadIdx.x;
  const int lane = tid & 31;
  const int wave = tid >> 5;
  const int lrow = lane & 15;
  const int half = lane >> 4;      // 0: lanes 0-15, 1: lanes 16-31
  const int wm   = wave >> 1;      // 0..3  (M direction)
  const int wn   = wave & 1;       // 0..1  (N direction)
  const int m0   = blockIdx.y * MBLK;
  const int n0   = blockIdx.x * NBLK;

  if (tid < MBLK) minbuf[tid] = 0xFFFFFFFFFFFFFFFFull;

  const v8f vzero = {0.f, 0.f, 0.f, 0.f, 0.f, 0.f, 0.f, 0.f};
  v8f acc[2][4] = {{vzero, vzero, vzero, vzero}, {vzero, vzero, vzero, vzero}};

  for (int kc = 0; kc < K_DIM; kc += KC) {
    // stage A: 128 x 32 (16 floats/thread, row-contiguous => coalesced)
#pragma unroll
    for (int i = 0; i < 16; ++i) {
      int l = tid + i * 256;
      int row = l >> 5, kk = l & 31;
      As[row * LDA + kk] = x[(size_t)(m0 + row) * K_DIM + kc + kk];
    }
    // stage B transposed: 32 x 128 -> Bs[n][k] (16 floats/thread, coalesced reads)
#pragma unroll
    for (int i = 0; i < 16; ++i) {
      int l = tid + i * 256;
      int kr = l >> 7, n = l & 127;
      Bs[n * LDA + kr] = kern[(size_t)(kc + kr) * N_DIM + n0 + n];
    }
    // prefetch next k-chunk into cache while this chunk computes (global_prefetch_b8)
    if (kc + KC < K_DIM) {
      int kn = kc + KC;
      // A next chunk: 128 rows x 32 floats = 256 cachelines; one 64B line per thread
      __builtin_prefetch(&x[(size_t)(m0 + (tid >> 1)) * K_DIM + kn + (tid & 1) * 16], 0, 3);
      // B next chunk: 32 rows x 128 floats = 256 cachelines; one 64B line per thread
      __builtin_prefetch(&kern[(size_t)(kn + (tid >> 3)) * N_DIM + n0 + (tid & 7) * 16], 0, 3);
    }
    __syncthreads();

#pragma unroll
    for (int ks = 0; ks < 8; ++ks) {
      // f32 16x16x4 fragment layout: half-wave h reads K = base+2h, base+2h+1
      int kb = ks * 4 + 2 * half;
      v2f a[2], b[4];
#pragma unroll
      for (int sm = 0; sm < 2; ++sm) {
        int ar = (wm * 32 + sm * 16 + lrow) * LDA + kb;
        a[sm].x = As[ar]; a[sm].y = As[ar + 1];
      }
#pragma unroll
      for (int sn = 0; sn < 4; ++sn) {
        int br = (wn * 64 + sn * 16 + lrow) * LDA + kb;
        b[sn].x = Bs[br]; b[sn].y = Bs[br + 1];
      }
#pragma unroll
      for (int sm = 0; sm < 2; ++sm)
#pragma unroll
        for (int sn = 0; sn < 4; ++sn)
          acc[sm][sn] = __builtin_amdgcn_wmma_f32_16x16x4_f32(
              false, a[sm], false, b[sn], (short)0, acc[sm][sn], false, false);
    }
    __syncthreads();
  }

  // epilogue: norms2 = x2[m] + k2[n] - 2*dot ; store + per-row packed argmin
  int ncol[4];
  float k2v[4];
#pragma unroll
  for (int sn = 0; sn < 4; ++sn) {
    ncol[sn] = n0 + wn * 64 + sn * 16 + lrow;
    k2v[sn]  = k2w[ncol[sn]];
  }
#pragma unroll
  for (int sm = 0; sm < 2; ++sm) {
#pragma unroll
    for (int r = 0; r < 8; ++r) {
      int m = m0 + wm * 32 + 16 * sm + r + 8 * half;  // C/D layout: M = r + 8*half
      float xs = x2w[m];
      unsigned long long c = 0xFFFFFFFFFFFFFFFFull;
#pragma unroll
      for (int sn = 0; sn < 4; ++sn) {
        float v = xs + k2v[sn] - 2.f * acc[sm][sn][r];
        out[(size_t)m * N_DIM + ncol[sn]] = v;
        unsigned long long cc = som_pack(v, (unsigned)ncol[sn]);
        if (cc < c) c = cc;
      }
#pragma unroll
      for (int sh = 1; sh <= 8; sh <<= 1) {   // reduce across 16 lanes of this half
        unsigned long long o = som_shfl_xor_u64(c, sh);
        if (o < c) c = o;
      }
      if (lrow == 0) atomicMin(&minbuf[m - m0], c);
    }
  }
  __syncthreads();
  if (tid < MBLK) atomicMin(&gmin[m0 + tid], minbuf[tid]);
}

// out[b,o] *= exp(-0.5*d2/sigma^2)/(sigma*sqrt(2*pi)), sigma=2, grid 128x128
__global__ void som_radial(float* __restrict__ out,
                           const unsigned long long* __restrict__ gmin) {
  unsigned gid = blockIdx.x * 256u + threadIdx.x;
  unsigned b = gid >> 14;          // /16384
  unsigned o = gid & 16383u;
  unsigned w = (unsigned)(gmin[b] & 0xFFFFFFFFull);
  int i  = (int)(o >> 7), j  = (int)(o & 127u);
  int wi = (int)(w >> 7), wj = (int)(w & 127u);
  int di = i - wi, dj = j - wj;
  float d2 = (float)(di * di + dj * dj);
  float rad = __expf(-0.125f * d2) * 0.19947114020071635f; // 1/(2*sqrt(2*pi))
  out[gid] = rad * out[gid];
}

extern "C" void kernel_launch(void* const* d_in, const int* in_sizes, int n_in,
                              void* d_out, int out_size, void* d_ws, size_t ws_size,
                              hipStream_t stream) {
  (void)in_sizes; (void)n_in; (void)out_size; (void)ws_size;
  const float* x    = (const float*)d_in[0];   // [4096, 256]
  const float* kern = (const float*)d_in[1];   // [256, 16384]
  float* out = (float*)d_out;                  // [4096, 16384]

  unsigned long long* gmin = (unsigned long long*)d_ws;              // 4096 * 8B
  float* x2w = (float*)((char*)d_ws + (size_t)M_DIM * 8);            // 4096 * 4B
  float* k2w = x2w + M_DIM;                                          // 16384 * 4B

  som_init_min<<<M_DIM / 256, 256, 0, stream>>>(gmin);
  som_x2<<<M_DIM / 8, 256, 0, stream>>>(x, x2w);
  som_k2<<<N_DIM / 256, 256, 0, stream>>>(kern, k2w);

  dim3 ggrid(N_DIM / NBLK, M_DIM / MBLK);  // 128 x 32
  som_gemm<<<ggrid, 256, 0, stream>>>(x, kern, x2w, k2w, out, gmin);

  som_radial<<<(unsigned)(((size_t)M_DIM * N_DIM) / 256), 256, 0, stream>>>(out, gmin);
}